// EmerGNN_65223373357148
// MI455X (gfx1250) — compile-verified
//
#include <hip/hip_runtime.h>
#include <math.h>

#define N_ENT   20000
#define BATCH   8
#define DIM     32
#define NEDGE   200000
#define NLAYER  2
#define ALL_RELN 115
#define EVAL_RELN 86
#define BB      16              // fused virtual batch: 8 head-dir + 8 tail-dir
#define TILE    (BB * DIM)      // 512 floats per entity

typedef __attribute__((ext_vector_type(2))) float v2f;
typedef __attribute__((ext_vector_type(4))) float v4f;
typedef __attribute__((ext_vector_type(8))) float v8f;

// ---------------- zero fill (vectorized) ----------------
__global__ void fill_zero(float* __restrict__ p, int n4) {
  int i = blockIdx.x * blockDim.x + threadIdx.x;
  if (i < n4) ((v4f*)p)[i] = (v4f){0.f, 0.f, 0.f, 0.f};
}

// ---------------- scatter initial embeddings into hid ----------------
// slot 0..7  : hid[head[b], b, :]   = ent_emb[head[b]]
// slot 8..15 : hid[tail[b], 8+b, :] = ent_emb[tail[b]]
__global__ void init_hid(const int* __restrict__ head, const int* __restrict__ tail,
                         const float* __restrict__ ent_emb, float* __restrict__ hid) {
  int slot = threadIdx.x >> 5;          // 0..15
  int d    = threadIdx.x & 31;
  int b    = slot & 7;
  int ent  = (slot & 8) ? tail[b] : head[b];
  hid[(size_t)ent * TILE + slot * DIM + d] = ent_emb[ent * DIM + d];
}

// ---------------- per-layer relation inputs: relw[l][b][r][d] ----------------
// rw = sigmoid(relu(ht @ W_rl.T + b_rl) @ W_ar.T + b_ar); relw = rw * rel_emb
__global__ void rel_weights(const int* __restrict__ head, const int* __restrict__ tail,
                            const float* __restrict__ ent_emb, const float* __restrict__ rel_emb,
                            const float* __restrict__ W_rl, const float* __restrict__ b_rl,
                            const float* __restrict__ W_ar, const float* __restrict__ b_ar,
                            float* __restrict__ relw) {
  int t = blockIdx.x * blockDim.x + threadIdx.x;
  if (t >= NLAYER * BATCH * ALL_RELN) return;
  int l   = t / (BATCH * ALL_RELN);
  int rem = t % (BATCH * ALL_RELN);
  int b   = rem / ALL_RELN;
  int r   = rem % ALL_RELN;
  const float* he = ent_emb + head[b] * DIM;
  const float* te = ent_emb + tail[b] * DIM;
  float h[5];
#pragma unroll
  for (int j = 0; j < 5; ++j) {
    const float* w = W_rl + (size_t)(l * 5 + j) * (2 * DIM);
    float a = b_rl[l * 5 + j];
#pragma unroll
    for (int k = 0; k < DIM; ++k) a += he[k] * w[k];
#pragma unroll
    for (int k = 0; k < DIM; ++k) a += te[k] * w[DIM + k];
    h[j] = fmaxf(a, 0.f);
  }
  const float* wa = W_ar + (size_t)(l * ALL_RELN + r) * 5;
  float z = b_ar[l * ALL_RELN + r];
#pragma unroll
  for (int j = 0; j < 5; ++j) z += h[j] * wa[j];
  float rw = 1.f / (1.f + expf(-z));
  const float* re  = rel_emb + (size_t)(l * ALL_RELN + r) * DIM;
  float*       out = relw + (((size_t)l * BATCH + b) * ALL_RELN + r) * DIM;
#pragma unroll
  for (int d = 0; d < DIM; ++d) out[d] = rw * re[d];
}

// ---------------- edge message pass: acc[src] += relw[b%8, rel] * hid[dst] ----------------
// one wave per edge; lane owns 16 contiguous floats of the 16x32 tile
__global__ void __launch_bounds__(256)
edge_spmm(const int* __restrict__ esrc, const int* __restrict__ edst,
          const int* __restrict__ erel, const float* __restrict__ relw_l,
          const float* __restrict__ hid, float* __restrict__ acc) {
  int e = blockIdx.x * 8 + (threadIdx.x >> 5);
  if (e >= NEDGE) return;
  int lane = threadIdx.x & 31;
  int s = esrc[e], d = edst[e], r = erel[e];
  int bb = lane >> 1;                       // virtual batch slot 0..15
  const float* hp = hid + (size_t)d * TILE + lane * 16;
  const float* rp = relw_l + ((size_t)(bb & 7) * ALL_RELN + r) * DIM + (lane & 1) * 16;
  float*       ap = acc + (size_t)s * TILE + lane * 16;
  v4f h[4], w[4];
#pragma unroll
  for (int i = 0; i < 4; ++i) { h[i] = *(const v4f*)(hp + 4 * i); w[i] = *(const v4f*)(rp + 4 * i); }
#pragma unroll
  for (int i = 0; i < 4; ++i) {
    __hip_atomic_fetch_add(ap + 4*i + 0, h[i].x * w[i].x, __ATOMIC_RELAXED, __HIP_MEMORY_SCOPE_AGENT);
    __hip_atomic_fetch_add(ap + 4*i + 1, h[i].y * w[i].y, __ATOMIC_RELAXED, __HIP_MEMORY_SCOPE_AGENT);
    __hip_atomic_fetch_add(ap + 4*i + 2, h[i].z * w[i].z, __ATOMIC_RELAXED, __HIP_MEMORY_SCOPE_AGENT);
    __hip_atomic_fetch_add(ap + 4*i + 3, h[i].w * w[i].w, __ATOMIC_RELAXED, __HIP_MEMORY_SCOPE_AGENT);
  }
}

// ---------------- per-entity linear: hid = relu(acc[16x32] @ W.T[32x32] + b) ----------------
// one wave per entity; 2 N-tiles x 8 K-steps of V_WMMA_F32_16X16X4_F32
__global__ void __launch_bounds__(256)
linear_relu(const float* __restrict__ acc, const float* __restrict__ W,
            const float* __restrict__ bias, float* __restrict__ hid) {
  int ent = blockIdx.x * 8 + (threadIdx.x >> 5);
  if (ent >= N_ENT) return;                 // wave-uniform; EXEC stays all-ones
  int lane = threadIdx.x & 31;
  int hl   = lane >> 4;                     // lane half: 0 or 1
  int m    = lane & 15;
  const float* A = acc + (size_t)ent * TILE + m * DIM;   // A row m (K-major)
  v2f af[8], b0f[8], b1f[8];
#pragma unroll
  for (int s = 0; s < 8; ++s) {
    // A frag: A[m, 4s + 2*hl .. +1] ; B frag (= W.T): W[n0+m, 4s + 2*hl .. +1]
    af[s]  = *(const v2f*)(A + 4 * s + 2 * hl);
    b0f[s] = *(const v2f*)(W + (size_t)(m)      * DIM + 4 * s + 2 * hl);
    b1f[s] = *(const v2f*)(W + (size_t)(16 + m) * DIM + 4 * s + 2 * hl);
  }
  v8f c0 = {}, c1 = {};
#pragma unroll
  for (int s = 0; s < 8; ++s) {
    c0 = __builtin_amdgcn_wmma_f32_16x16x4_f32(false, af[s], false, b0f[s], (short)0, c0, false, false);
    c1 = __builtin_amdgcn_wmma_f32_16x16x4_f32(false, af[s], false, b1f[s], (short)0, c1, false, false);
  }
  float bias0 = bias[m], bias1 = bias[16 + m];
  float* O = hid + (size_t)ent * TILE;
#pragma unroll
  for (int v = 0; v < 8; ++v) {
    int row = v + 8 * hl;                   // C/D layout: lanes 16..31 hold M = v+8
    O[row * DIM + m]      = fmaxf(c0[v] + bias0, 0.f);
    O[row * DIM + 16 + m] = fmaxf(c1[v] + bias1, 0.f);
  }
}

// ---------------- final readout: scores = [he, hh, th, te] @ Wr.T + br ----------------
__global__ void readout(const int* __restrict__ head, const int* __restrict__ tail,
                        const float* __restrict__ ent_emb, const float* __restrict__ hid,
                        const float* __restrict__ Wr, const float* __restrict__ br,
                        float* __restrict__ out) {
  int t = blockIdx.x * blockDim.x + threadIdx.x;
  if (t >= BATCH * EVAL_RELN) return;
  int b = t / EVAL_RELN, r = t % EVAL_RELN;
  int hb = head[b], tb = tail[b];
  const float* w  = Wr + (size_t)r * (4 * DIM);
  const float* he = ent_emb + hb * DIM;
  const float* hh = hid + (size_t)hb * TILE + (8 + b) * DIM;  // tail->head pass read at head
  const float* th = hid + (size_t)tb * TILE + b * DIM;        // head->tail pass read at tail
  const float* te = ent_emb + tb * DIM;
  float sc = br[r];
#pragma unroll
  for (int k = 0; k < DIM; ++k) sc += w[k] * he[k];
#pragma unroll
  for (int k = 0; k < DIM; ++k) sc += w[DIM + k] * hh[k];
#pragma unroll
  for (int k = 0; k < DIM; ++k) sc += w[2 * DIM + k] * th[k];
#pragma unroll
  for (int k = 0; k < DIM; ++k) sc += w[3 * DIM + k] * te[k];
  out[t] = sc;
}

extern "C" void kernel_launch(void* const* d_in, const int* in_sizes, int n_in,
                              void* d_out, int out_size, void* d_ws, size_t ws_size,
                              hipStream_t stream) {
  (void)in_sizes; (void)n_in; (void)out_size; (void)ws_size;
  const int*   head    = (const int*)d_in[0];
  const int*   tail    = (const int*)d_in[1];
  const int*   esrc    = (const int*)d_in[2];
  const int*   edst    = (const int*)d_in[3];
  const int*   erel    = (const int*)d_in[4];
  const float* ent_emb = (const float*)d_in[5];
  const float* rel_emb = (const float*)d_in[6];
  const float* W_lin   = (const float*)d_in[7];
  const float* b_lin   = (const float*)d_in[8];
  const float* W_rl    = (const float*)d_in[9];
  const float* b_rl    = (const float*)d_in[10];
  const float* W_ar    = (const float*)d_in[11];
  const float* b_ar    = (const float*)d_in[12];
  const float* Wr      = (const float*)d_in[13];
  const float* br      = (const float*)d_in[14];

  char*  ws     = (char*)d_ws;
  size_t hbytes = (size_t)N_ENT * TILE * sizeof(float);   // ~41 MB, L2-resident
  float* hid  = (float*)(ws);
  float* acc  = (float*)(ws + hbytes);
  float* relw = (float*)(ws + 2 * hbytes);                // [L][8][115][32]

  int n4 = N_ENT * TILE / 4;
  fill_zero<<<(n4 + 255) / 256, 256, 0, stream>>>(hid, n4);
  init_hid<<<1, 512, 0, stream>>>(head, tail, ent_emb, hid);
  int nrw = NLAYER * BATCH * ALL_RELN;
  rel_weights<<<(nrw + 255) / 256, 256, 0, stream>>>(head, tail, ent_emb, rel_emb,
                                                     W_rl, b_rl, W_ar, b_ar, relw);
  for (int l = 0; l < NLAYER; ++l) {
    fill_zero<<<(n4 + 255) / 256, 256, 0, stream>>>(acc, n4);
    edge_spmm<<<NEDGE / 8, 256, 0, stream>>>(esrc, edst, erel,
                                             relw + (size_t)l * BATCH * ALL_RELN * DIM,
                                             hid, acc);
    linear_relu<<<N_ENT / 8, 256, 0, stream>>>(acc, W_lin + (size_t)l * DIM * DIM,
                                               b_lin + l * DIM, hid);
  }
  readout<<<(BATCH * EVAL_RELN + 255) / 256, 256, 0, stream>>>(head, tail, ent_emb, hid,
                                                               Wr, br, (float*)d_out);
}